// TransportKernel_73126113182126
// MI455X (gfx1250) — compile-verified
//
#include <hip/hip_runtime.h>

#define N        4096
#define NT       256          // 16x16 tiles per dimension
#define NUGGETF  0.001f
#define REG_LAM  0.01f
#define CG_ITERS 128

typedef __attribute__((ext_vector_type(2))) float v2f;
typedef __attribute__((ext_vector_type(8))) float v8f;
typedef __attribute__((ext_vector_type(4))) int   v4i;

__device__ __forceinline__ float wave_sum32(float v) {
#pragma unroll
  for (int m = 16; m >= 1; m >>= 1) v += __shfl_xor(v, m, 32);
  return v;
}

// --------------------------------------------------------------------------
// Row squared norms of X, Y, Z.
__global__ __launch_bounds__(256) void norms_kernel(
    const float4* __restrict__ X4, const float4* __restrict__ Y4,
    const float4* __restrict__ Z4, float* __restrict__ xn,
    float* __restrict__ yn, float* __restrict__ zn) {
  int i = blockIdx.x * 256 + threadIdx.x;
  float4 x = X4[i]; xn[i] = x.x * x.x + x.y * x.y + x.z * x.z + x.w * x.w;
  float4 y = Y4[i]; yn[i] = y.x * y.x + y.y * y.y + y.z * y.z + y.w * y.w;
  float4 z = Z4[i]; zn[i] = z.x * z.x + z.y * z.y + z.z * z.z + z.w * z.w;
}

// --------------------------------------------------------------------------
// Build fit_kXX + nugget*I into ws (64 MB, L2-resident).
// One wave per 16x16 tile; cross term via V_WMMA_F32_16X16X4_F32 (K = D = 4).
__global__ __launch_bounds__(256) void build_kxx_kernel(
    const float4* __restrict__ X4, const float* __restrict__ xn,
    float* __restrict__ K) {
  const int lane = threadIdx.x & 31;
  const int wave = threadIdx.x >> 5;
  const int tile = blockIdx.x * 8 + wave;
  const int ti = tile >> 8, tj = tile & 255;
  const int half = lane >> 4, l15 = lane & 15;

  // A (16x4 f32): lanes 0-15 M=0..15 {K0,K1}; lanes 16-31 {K2,K3}
  float4 xa = X4[ti * 16 + l15];
  v2f a; a[0] = half ? xa.z : xa.x; a[1] = half ? xa.w : xa.y;
  // B (4x16 f32) = X^T tile: v0 = rows K0/K2, v1 = rows K1/K3, N = l15
  const int col = tj * 16 + l15;
  float4 xb = X4[col];
  v2f b; b[0] = half ? xb.z : xb.x; b[1] = half ? xb.w : xb.y;

  v8f c = {};
  c = __builtin_amdgcn_wmma_f32_16x16x4_f32(false, a, false, b, (short)0, c,
                                            false, false);

  const float xn_col = xn[col];
  const float4 r0 = ((const float4*)xn)[ti * 4 + half * 2];
  const float4 r1 = ((const float4*)xn)[ti * 4 + half * 2 + 1];
  const float rn[8] = {r0.x, r0.y, r0.z, r0.w, r1.x, r1.y, r1.z, r1.w};

#pragma unroll
  for (int r = 0; r < 8; ++r) {
    const int row = ti * 16 + half * 8 + r;
    float sq = fmaxf(rn[r] + xn_col - 2.0f * c[r], 0.0f);
    float k = __expf(-0.5f * sq);            // sigma = l = 1
    if (row == col) k += NUGGETF;
    K[(size_t)row * N + col] = k;
  }
}

// --------------------------------------------------------------------------
// loss_fit = sum(kZZ^2) - 2*sum(kZY^2), fused: 2 WMMAs + exp + wave reduce.
__global__ __launch_bounds__(256) void gram_loss_fit_kernel(
    const float4* __restrict__ Z4, const float4* __restrict__ Y4,
    const float* __restrict__ zn, const float* __restrict__ yn,
    float* __restrict__ accFit) {
  const int lane = threadIdx.x & 31;
  const int wave = threadIdx.x >> 5;
  const int tile = blockIdx.x * 8 + wave;
  const int ti = tile >> 8, tj = tile & 255;
  const int half = lane >> 4, l15 = lane & 15;

  float4 za = Z4[ti * 16 + l15];
  v2f a; a[0] = half ? za.z : za.x; a[1] = half ? za.w : za.y;

  const int col = tj * 16 + l15;
  float4 zb = Z4[col];
  v2f bz; bz[0] = half ? zb.z : zb.x; bz[1] = half ? zb.w : zb.y;
  float4 yb = Y4[col];
  v2f by; by[0] = half ? yb.z : yb.x; by[1] = half ? yb.w : yb.y;

  v8f czz = {};
  czz = __builtin_amdgcn_wmma_f32_16x16x4_f32(false, a, false, bz, (short)0,
                                              czz, false, false);
  v8f czy = {};
  czy = __builtin_amdgcn_wmma_f32_16x16x4_f32(false, a, false, by, (short)0,
                                              czy, false, false);

  const float zn_col = zn[col];
  const float yn_col = yn[col];
  const float4 r0 = ((const float4*)zn)[ti * 4 + half * 2];
  const float4 r1 = ((const float4*)zn)[ti * 4 + half * 2 + 1];
  const float rn[8] = {r0.x, r0.y, r0.z, r0.w, r1.x, r1.y, r1.z, r1.w};

  float part = 0.0f;
#pragma unroll
  for (int r = 0; r < 8; ++r) {
    float szz = fmaxf(rn[r] + zn_col - 2.0f * czz[r], 0.0f);
    float kzz = __expf(-0.5f * szz);
    float szy = fmaxf(rn[r] + yn_col - 2.0f * czy[r], 0.0f);
    float kzy = __expf(-0.5f * szy);
    part += kzz * kzz - 2.0f * kzy * kzy;
  }
  part = wave_sum32(part);
  if (lane == 0) atomicAdd(accFit, part);
}

// --------------------------------------------------------------------------
// CG solve of K w = Z (4 RHS fused). Matvec: p staged into LDS via CDNA5
// async global->LDS DMA (ASYNCcnt path, no VGPR round-trip), K streamed from
// L2 with software prefetch; 16 rows per 256-thread block.
__global__ __launch_bounds__(256) void matvec_kernel(
    const float* __restrict__ K, const float4* __restrict__ p,
    float4* __restrict__ Ap) {
  __shared__ float4 pl[N];                      // 64 KB of 320 KB WGP LDS
  // Async bulk copy: each lane moves 16 B per op straight into LDS.
  for (int j = threadIdx.x; j < N; j += 256) {
    __builtin_amdgcn_global_load_async_to_lds_b128((v4i*)&p[j], (v4i*)&pl[j],
                                                   0, 0);
  }
  asm volatile("s_wait_asynccnt 0x0" ::: "memory");
  __syncthreads();

  const int tx = threadIdx.x & 15;
  const int ty = threadIdx.x >> 4;
  const int row = blockIdx.x * 16 + ty;
  const float4* Kr = (const float4*)(K + (size_t)row * N);
  float4 acc = make_float4(0.f, 0.f, 0.f, 0.f);
  for (int j4 = tx; j4 < N / 4; j4 += 16) {
    __builtin_prefetch(&Kr[j4 + 16], 0, 3);     // global_prefetch_b8
    float4 kv = Kr[j4];
    float4 p0 = pl[4 * j4 + 0], p1 = pl[4 * j4 + 1];
    float4 p2 = pl[4 * j4 + 2], p3 = pl[4 * j4 + 3];
    acc.x += kv.x * p0.x + kv.y * p1.x + kv.z * p2.x + kv.w * p3.x;
    acc.y += kv.x * p0.y + kv.y * p1.y + kv.z * p2.y + kv.w * p3.y;
    acc.z += kv.x * p0.z + kv.y * p1.z + kv.z * p2.z + kv.w * p3.z;
    acc.w += kv.x * p0.w + kv.y * p1.w + kv.z * p2.w + kv.w * p3.w;
  }
#pragma unroll
  for (int m = 8; m >= 1; m >>= 1) {
    acc.x += __shfl_xor(acc.x, m, 32);
    acc.y += __shfl_xor(acc.y, m, 32);
    acc.z += __shfl_xor(acc.z, m, 32);
    acc.w += __shfl_xor(acc.w, m, 32);
  }
  if (tx == 0) Ap[row] = acc;
}

__global__ __launch_bounds__(256) void cg_init_kernel(
    const float4* __restrict__ Z4, float4* __restrict__ w,
    float4* __restrict__ r, float4* __restrict__ p, float* __restrict__ rsold) {
  int i = blockIdx.x * 256 + threadIdx.x;
  float4 z = Z4[i];
  w[i] = make_float4(0.f, 0.f, 0.f, 0.f);
  r[i] = z; p[i] = z;
  float s0 = wave_sum32(z.x * z.x), s1 = wave_sum32(z.y * z.y);
  float s2 = wave_sum32(z.z * z.z), s3 = wave_sum32(z.w * z.w);
  if ((threadIdx.x & 31) == 0) {
    atomicAdd(&rsold[0], s0); atomicAdd(&rsold[1], s1);
    atomicAdd(&rsold[2], s2); atomicAdd(&rsold[3], s3);
  }
}

__global__ void cg_prepare_kernel(float* rsold, float* rsnew, float* pAp,
                                  int first) {
  int d = threadIdx.x;
  if (d < 4) {
    if (!first) rsold[d] = rsnew[d];
    rsnew[d] = 0.f;
    pAp[d] = 0.f;
  }
}

__global__ __launch_bounds__(256) void dot_pAp_kernel(
    const float4* __restrict__ p, const float4* __restrict__ Ap,
    float* __restrict__ pAp) {
  int i = blockIdx.x * 256 + threadIdx.x;
  float4 pv = p[i], av = Ap[i];
  float s0 = wave_sum32(pv.x * av.x), s1 = wave_sum32(pv.y * av.y);
  float s2 = wave_sum32(pv.z * av.z), s3 = wave_sum32(pv.w * av.w);
  if ((threadIdx.x & 31) == 0) {
    atomicAdd(&pAp[0], s0); atomicAdd(&pAp[1], s1);
    atomicAdd(&pAp[2], s2); atomicAdd(&pAp[3], s3);
  }
}

__global__ __launch_bounds__(256) void cg_update1_kernel(
    float4* __restrict__ w, float4* __restrict__ r,
    const float4* __restrict__ p, const float4* __restrict__ Ap,
    const float* __restrict__ rsold, const float* __restrict__ pAp,
    float* __restrict__ rsnew) {
  int i = blockIdx.x * 256 + threadIdx.x;
  float a0 = rsold[0] / fmaxf(pAp[0], 1e-30f);
  float a1 = rsold[1] / fmaxf(pAp[1], 1e-30f);
  float a2 = rsold[2] / fmaxf(pAp[2], 1e-30f);
  float a3 = rsold[3] / fmaxf(pAp[3], 1e-30f);
  float4 pv = p[i], av = Ap[i], wv = w[i], rv = r[i];
  wv.x += a0 * pv.x; wv.y += a1 * pv.y; wv.z += a2 * pv.z; wv.w += a3 * pv.w;
  rv.x -= a0 * av.x; rv.y -= a1 * av.y; rv.z -= a2 * av.z; rv.w -= a3 * av.w;
  w[i] = wv; r[i] = rv;
  float s0 = wave_sum32(rv.x * rv.x), s1 = wave_sum32(rv.y * rv.y);
  float s2 = wave_sum32(rv.z * rv.z), s3 = wave_sum32(rv.w * rv.w);
  if ((threadIdx.x & 31) == 0) {
    atomicAdd(&rsnew[0], s0); atomicAdd(&rsnew[1], s1);
    atomicAdd(&rsnew[2], s2); atomicAdd(&rsnew[3], s3);
  }
}

__global__ __launch_bounds__(256) void cg_update2_kernel(
    float4* __restrict__ p, const float4* __restrict__ r,
    const float* __restrict__ rsold, const float* __restrict__ rsnew) {
  int i = blockIdx.x * 256 + threadIdx.x;
  float b0 = rsnew[0] / fmaxf(rsold[0], 1e-30f);
  float b1 = rsnew[1] / fmaxf(rsold[1], 1e-30f);
  float b2 = rsnew[2] / fmaxf(rsold[2], 1e-30f);
  float b3 = rsnew[3] / fmaxf(rsold[3], 1e-30f);
  float4 rv = r[i], pv = p[i];
  pv.x = rv.x + b0 * pv.x; pv.y = rv.y + b1 * pv.y;
  pv.z = rv.z + b2 * pv.z; pv.w = rv.w + b3 * pv.w;
  p[i] = pv;
}

__global__ __launch_bounds__(256) void reg_dot_kernel(
    const float4* __restrict__ Z4, const float4* __restrict__ w,
    float* __restrict__ accReg) {
  int i = blockIdx.x * 256 + threadIdx.x;
  float4 z = Z4[i], wv = w[i];
  float part = z.x * wv.x + z.y * wv.y + z.z * wv.z + z.w * wv.w;
  part = wave_sum32(part);
  if ((threadIdx.x & 31) == 0) atomicAdd(accReg, part);
}

__global__ void clear_kernel(float* s) {
  if (threadIdx.x < 16) s[threadIdx.x] = 0.f;
}

__global__ void finalize_kernel(const float* accFit, const float* accReg,
                                float* out) {
  if (threadIdx.x == 0) out[0] = accFit[0] + REG_LAM * accReg[0];
}

// --------------------------------------------------------------------------
extern "C" void kernel_launch(void* const* d_in, const int* in_sizes, int n_in,
                              void* d_out, int out_size, void* d_ws,
                              size_t ws_size, hipStream_t stream) {
  const float4* X4 = (const float4*)d_in[0];
  const float4* Y4 = (const float4*)d_in[1];
  const float4* Z4 = (const float4*)d_in[2];
  float* out = (float*)d_out;

  // workspace layout
  float* K = (float*)d_ws;                         // 64 MB (L2-resident)
  float* f = K + (size_t)N * N;
  float4* w  = (float4*)f;  f += (size_t)N * 4;
  float4* r  = (float4*)f;  f += (size_t)N * 4;
  float4* p  = (float4*)f;  f += (size_t)N * 4;
  float4* Ap = (float4*)f;  f += (size_t)N * 4;
  float* xn = f; f += N;
  float* yn = f; f += N;
  float* zn = f; f += N;
  float* scal  = f;          // 16 scalars
  float* rsold = scal + 0;
  float* rsnew = scal + 4;
  float* pAp   = scal + 8;
  float* accFit = scal + 12;
  float* accReg = scal + 13;

  clear_kernel<<<1, 32, 0, stream>>>(scal);
  norms_kernel<<<N / 256, 256, 0, stream>>>(X4, Y4, Z4, xn, yn, zn);
  build_kxx_kernel<<<(NT * NT) / 8, 256, 0, stream>>>(X4, xn, K);
  gram_loss_fit_kernel<<<(NT * NT) / 8, 256, 0, stream>>>(Z4, Y4, zn, yn,
                                                          accFit);
  cg_init_kernel<<<N / 256, 256, 0, stream>>>(Z4, w, r, p, rsold);
  for (int it = 0; it < CG_ITERS; ++it) {
    cg_prepare_kernel<<<1, 32, 0, stream>>>(rsold, rsnew, pAp, it == 0);
    matvec_kernel<<<N / 16, 256, 0, stream>>>(K, p, Ap);
    dot_pAp_kernel<<<N / 256, 256, 0, stream>>>(p, Ap, pAp);
    cg_update1_kernel<<<N / 256, 256, 0, stream>>>(w, r, p, Ap, rsold, pAp,
                                                   rsnew);
    cg_update2_kernel<<<N / 256, 256, 0, stream>>>(p, r, rsold, rsnew);
  }
  reg_dot_kernel<<<N / 256, 256, 0, stream>>>(Z4, w, accReg);
  finalize_kernel<<<1, 32, 0, stream>>>(accFit, accReg, out);
}